// GCN_91285234909358
// MI455X (gfx1250) — compile-verified
//
#include <hip/hip_runtime.h>
#include <hip/hip_bf16.h>

typedef float v2f __attribute__((ext_vector_type(2)));
typedef float v8f __attribute__((ext_vector_type(8)));
typedef int   v4i __attribute__((ext_vector_type(4)));

typedef __attribute__((address_space(1))) v4i* gas_v4i_ptr;   // global
typedef __attribute__((address_space(3))) v4i* las_v4i_ptr;   // LDS

#define IN_CH 256

#if __has_builtin(__builtin_amdgcn_global_load_async_to_lds_b128)
#define HAVE_ASYNC_LDS 1
#else
#define HAVE_ASYNC_LDS 0
#endif

// ---------------------------------------------------------------- utilities

__global__ void k_zero_f32(float* __restrict__ p, int n) {
    int t = blockIdx.x * blockDim.x + threadIdx.x;
    if (t < n) p[t] = 0.0f;
}

__global__ void k_deg_count(const int* __restrict__ dst, int* __restrict__ deg, int E) {
    int t = blockIdx.x * blockDim.x + threadIdx.x;
    if (t < E) atomicAdd(&deg[dst[t]], 1);
}

// dinv[i] = rsqrt(edge_in_degree + 1 self loop); always >= 1 so no zero guard needed
__global__ void k_dinv(const int* __restrict__ deg, float* __restrict__ dinv, int n) {
    int t = blockIdx.x * blockDim.x + threadIdx.x;
    if (t < n) dinv[t] = rsqrtf((float)(deg[t] + 1));
}

// ---------------------------------------------------------------- fp32 WMMA GEMM
// H[M,N] = X[M,K] @ W[K,N], row-major. M % 16 == 0, K in {64,128,256}, N = 16*wavesPerBlock.
// Block computes a 16-row stripe across all N columns; the 16xK A-tile (one contiguous
// 16*K-float span of X) is staged into LDS once (async DMA when available), then each
// wave runs the K-loop with ds_load_b64 A-fragments and global B-fragments.
// LDS row stride = K+4 floats: dword addr = 4*l16 + k + 2*half (mod 64) -> conflict-free.

#define LDS_PAD 4
#define LDS_MAX (16 * (256 + LDS_PAD))

__global__ __launch_bounds__(256) void k_gemm_wmma_f32(
        const float* __restrict__ X, const float* __restrict__ W,
        float* __restrict__ H, int K, int N, int lgCPR /* = lg2(K/4) */) {
    __shared__ float lds[LDS_MAX];

    const int wave = threadIdx.x >> 5;
    const int lane = threadIdx.x & 31;
    const int half = lane >> 4;          // 0: lanes 0-15, 1: lanes 16-31
    const int l16  = lane & 15;
    const int row0 = blockIdx.x << 4;
    const int ldl  = K + LDS_PAD;        // LDS row stride (floats)

    // ---- stage A tile: 16*K floats, contiguous in X starting at row0*K ----
    {
        const int chunks = (16 * K) >> 2;              // 16B chunks
        const int cpr_m1 = (1 << lgCPR) - 1;           // chunks per row - 1
        const char* gsrc = (const char*)(X + (size_t)row0 * K);
        for (int c = threadIdx.x; c < chunks; c += blockDim.x) {
            const int r   = c >> lgCPR;
            const int col = c & cpr_m1;
            const int lofs = (r * ldl + (col << 2)) * 4;   // bytes
#if HAVE_ASYNC_LDS
            __builtin_amdgcn_global_load_async_to_lds_b128(
                (gas_v4i_ptr)(gsrc + ((size_t)c << 4)),
                (las_v4i_ptr)((char*)lds + lofs), 0, 0);
#else
            *(float4*)((char*)lds + lofs) = *(const float4*)(gsrc + ((size_t)c << 4));
#endif
        }
    }
#if HAVE_ASYNC_LDS
#if __has_builtin(__builtin_amdgcn_s_wait_asynccnt)
    __builtin_amdgcn_s_wait_asynccnt(0);
#else
    asm volatile("s_wait_asynccnt 0" ::: "memory");
#endif
#endif
    __syncthreads();

    // ---- per-wave 16x16 tile ----
    const int col0 = wave << 4;
    if (col0 >= N) return;               // wave-uniform: EXEC all-ones for WMMA

    const float* ap = &lds[l16 * ldl + 2 * half];       // A: row l16, K = k + 2*half
    const float* bp = W + (size_t)(2 * half) * N + col0 + l16;  // B: col n, K = k + 2*half
    const int bstep = 4 * N;

    v8f acc = {};
    #pragma unroll 4
    for (int k = 0; k < K; k += 4) {
        v2f a, b;
        a.x = ap[0];
        a.y = ap[1];
        b.x = bp[0];
        b.y = bp[N];
        acc = __builtin_amdgcn_wmma_f32_16x16x4_f32(
                false, a, false, b, (short)0, acc, false, false);
        ap += 4;
        bp += bstep;
    }
    // D layout: VGPR r holds row M = r + 8*half
    float* __restrict__ hp = H + (size_t)(row0 + 8 * half) * N + col0 + l16;
    #pragma unroll
    for (int r = 0; r < 8; ++r) hp[(size_t)r * N] = acc[r];
}

// ---------------------------------------------------------------- aggregation
// AGG[i][c] = H[i][c] * dinv[i]^2 (self-loop message) + bias[c]
__global__ void k_agg_init(const float* __restrict__ H, const float* __restrict__ dinv,
                           const float* __restrict__ bias, float* __restrict__ AGG,
                           long long total, int lgC, int cmask) {
    long long t = blockIdx.x * (long long)blockDim.x + threadIdx.x;
    if (t >= total) return;
    const int i = (int)(t >> lgC);
    const int c = (int)t & cmask;
    const float dv = dinv[i];
    AGG[t] = H[t] * dv * dv + bias[c];
}

// AGG[dst] += H[src] * dinv[src]*dinv[dst]; one lane handles (edge, 4 channels)
__global__ void k_scatter(const float* __restrict__ H, const float* __restrict__ dinv,
                          const int* __restrict__ src, const int* __restrict__ dst,
                          float* __restrict__ AGG, int E, int C, int lgCg, int cgmask) {
    long long t = blockIdx.x * (long long)blockDim.x + threadIdx.x;
    const int e = (int)(t >> lgCg);
    if (e >= E) return;
    const int g = ((int)t & cgmask) << 2;   // channel group start
    const int s = src[e], d = dst[e];
    const float nrm = dinv[s] * dinv[d];
    const float4 hv = *(const float4*)(H + (size_t)s * C + g);
    float* __restrict__ ap = AGG + (size_t)d * C + g;
    atomicAdd(ap + 0, hv.x * nrm);
    atomicAdd(ap + 1, hv.y * nrm);
    atomicAdd(ap + 2, hv.z * nrm);
    atomicAdd(ap + 3, hv.w * nrm);
}

// ---------------------------------------------------------------- batch norm
// sums[0..C-1] = sum_i X[i][c]; sums[C..2C-1] = sum_i X[i][c]^2. blockDim.x == C.
__global__ void k_bn_stats(const float* __restrict__ X, float* __restrict__ sums,
                           int n, int C) {
    const int c = threadIdx.x;
    float s = 0.0f, q = 0.0f;
    for (int r = blockIdx.x; r < n; r += gridDim.x) {
        const float v = X[(size_t)r * C + c];
        s += v;
        q += v * v;
    }
    atomicAdd(&sums[c], s);
    atomicAdd(&sums[C + c], q);
}

// in-place: X = relu((X - mu) * rsqrt(var + eps) * gamma + beta)
__global__ void k_bn_apply(float* __restrict__ X, const float* __restrict__ sums,
                           const float* __restrict__ gamma, const float* __restrict__ beta,
                           long long total, float inv_n, int C, int cmask) {
    long long t = blockIdx.x * (long long)blockDim.x + threadIdx.x;
    if (t >= total) return;
    const int c = (int)t & cmask;
    const float mu  = sums[c] * inv_n;
    const float var = sums[C + c] * inv_n - mu * mu;
    const float g   = gamma[c] * rsqrtf(var + 1e-5f);
    const float v   = (X[t] - mu) * g + beta[c];
    X[t] = v > 0.0f ? v : 0.0f;
}

// ---------------------------------------------------------------- host side

static inline int ilg2(int x) { int l = 0; while ((1 << l) < x) ++l; return l; }

static void run_layer(const float* Xin, const float* W, const float* bias,
                      const float* gamma, const float* beta, bool do_bn,
                      float* Hbuf, float* Agg, float* bns,
                      const int* src, const int* dst, const float* dinv,
                      int n, int E, int K, int C, hipStream_t stream) {
    // GEMM: H = Xin @ W
    {
        const int waves = C / 16;                  // 8 (C=128) or 4 (C=64)
        dim3 grid(n / 16, 1, 1);
        dim3 block(waves * 32, 1, 1);
        k_gemm_wmma_f32<<<grid, block, 0, stream>>>(Xin, W, Hbuf, K, C, ilg2(K / 4));
    }
    const int lgC = ilg2(C);
    const long long totalNC = (long long)n * C;
    // self-loop + bias (fully initializes Agg)
    {
        const long long blocks = (totalNC + 255) / 256;
        k_agg_init<<<(unsigned)blocks, 256, 0, stream>>>(Hbuf, dinv, bias, Agg,
                                                         totalNC, lgC, C - 1);
    }
    // edge scatter
    {
        const int cg = C >> 2;
        const int lgCg = lgC - 2;
        const long long threads = (long long)E * cg;
        const long long blocks = (threads + 255) / 256;
        k_scatter<<<(unsigned)blocks, 256, 0, stream>>>(Hbuf, dinv, src, dst, Agg,
                                                        E, C, lgCg, cg - 1);
    }
    if (do_bn) {
        k_zero_f32<<<1, 256, 0, stream>>>(bns, 2 * C);
        k_bn_stats<<<512, C, 0, stream>>>(Agg, bns, n, C);
        const long long blocks = (totalNC + 255) / 256;
        k_bn_apply<<<(unsigned)blocks, 256, 0, stream>>>(Agg, bns, gamma, beta,
                                                         totalNC, 1.0f / n, C, C - 1);
    }
}

extern "C" void kernel_launch(void* const* d_in, const int* in_sizes, int n_in,
                              void* d_out, int out_size, void* d_ws, size_t ws_size,
                              hipStream_t stream) {
    const float* x   = (const float*)d_in[0];
    const int*   ei  = (const int*)d_in[1];
    const float* W1  = (const float*)d_in[2];
    const float* b1  = (const float*)d_in[3];
    const float* g1  = (const float*)d_in[4];
    const float* be1 = (const float*)d_in[5];
    const float* W2  = (const float*)d_in[6];
    const float* b2  = (const float*)d_in[7];
    const float* g2  = (const float*)d_in[8];
    const float* be2 = (const float*)d_in[9];
    const float* W3  = (const float*)d_in[10];
    const float* b3  = (const float*)d_in[11];

    const int n = in_sizes[0] / IN_CH;      // 100000 nodes (divisible by 16)
    const int E = in_sizes[1] / 2;          // 1.6M edges
    const int* src = ei;
    const int* dst = ei + E;

    // workspace layout
    char* ws = (char*)d_ws;
    int*   deg  = (int*)(ws + 0);                        // n ints      (400 KB)
    float* dinv = (float*)(ws + 400000);                 // n floats    (400 KB)
    float* bns  = (float*)(ws + 800000);                 // 256 floats
    float* bufA = (float*)(ws + 1048576);                // 51.2 MB (H buffer)
    float* bufB = (float*)(ws + 1048576 + 51200000);     // 51.2 MB (AGG buffer)

    // degrees -> dinv
    k_zero_f32<<<(n + 255) / 256, 256, 0, stream>>>((float*)deg, n);
    k_deg_count<<<(E + 255) / 256, 256, 0, stream>>>(dst, deg, E);
    k_dinv<<<(n + 255) / 256, 256, 0, stream>>>(deg, dinv, n);

    // layer 1: 256 -> 128, BN+ReLU (bufB holds activations afterwards)
    run_layer(x, W1, b1, g1, be1, true, bufA, bufB, bns,
              src, dst, dinv, n, E, 256, 128, stream);
    // layer 2: 128 -> 64, BN+ReLU (gemm reads bufB, agg overwrites bufB after gemm completes)
    run_layer(bufB, W2, b2, g2, be2, true, bufA, bufB, bns,
              src, dst, dinv, n, E, 128, 64, stream);
    // layer 3: 64 -> 64, no BN, aggregate straight into d_out
    run_layer(bufB, W3, b3, nullptr, nullptr, false, bufA, (float*)d_out, bns,
              src, dst, dinv, n, E, 64, 64, stream);
}